// StructuralImportanceAttentionPure_15040975470962
// MI455X (gfx1250) — compile-verified
//
#include <hip/hip_runtime.h>
#include <math.h>

// CDNA5 / gfx1250. wave32. WMMA f32 16x16x4 path for both GEMMs.

typedef __attribute__((ext_vector_type(2))) float v2f;
typedef __attribute__((ext_vector_type(8))) float v8f;

#define HIDDEN 256
#define PROJ   64
#define PROJ2  128          // keys | vals concatenated per edge
#define RPB    64           // rows per block (4 row-tiles of 16)
#define XSTRIDE 260         // 256 + 4 pad -> stride mod 64banks = 4 -> conflict-free
#define HSTRIDE 68          // 64 + 4 pad

// ---------------------------------------------------------------------------
// K1: gathered projection GEMM + fused segment-sum of keys & member count.
// proj[e][0:64)=keys, [64:128)=vals.
// One block = 64 gathered rows; 8 waves x one 16-col tile of 128 cols,
// each wave keeps 4 row-tile accumulators so every B fragment feeds 4 WMMAs.
// ---------------------------------------------------------------------------
__global__ __launch_bounds__(256)
void k_project(const float* __restrict__ node_feats,
               const float* __restrict__ W_k,
               const float* __restrict__ W_v,
               const int*   __restrict__ node_idx,
               const int*   __restrict__ he_idx,
               float* __restrict__ proj,
               float* __restrict__ ksum,
               float* __restrict__ cnt,
               int nnz)
{
    __shared__ float xs[RPB * XSTRIDE];   // ~65 KB (gfx1250: 320 KB/WGP)
    const int e0 = blockIdx.x * RPB;
    const int t  = threadIdx.x;

    // Fused member count (1 atomic per edge).
    if (t < RPB) {
        const int e = e0 + t;
        if (e < nnz) atomicAdd(&cnt[he_idx[e]], 1.0f);
    }

    // Stage 64 gathered rows: 4 threads per row, 64 floats each (float4).
    {
        const int r   = t >> 2;
        const int c0  = (t & 3) * 64;
        const int e   = e0 + r;
        const int row = (e < nnz) ? node_idx[e] : 0;   // clamp keeps EXEC full
        const float* src = node_feats + (long)row * HIDDEN + c0;
        float* dst = xs + r * XSTRIDE + c0;
        #pragma unroll
        for (int i = 0; i < 64; i += 4)
            *(float4*)(dst + i) = *(const float4*)(src + i);
    }
    __syncthreads();

    const int wave = t >> 5;             // 0..7 -> col tile n0 = wave*16
    const int lane = t & 31;
    const int mn   = lane & 15;          // M for A-frag, N for B-frag
    const int kb   = (lane >> 4) * 2;    // K sub-offset: 0 or 2
    const int mh   = (lane >> 4) * 8;    // D row-half offset
    const int n0   = wave * 16;

    // B[k][n] = W[n][k] where W = concat(W_k, W_v) rows.
    const int nrow = n0 + mn;
    const float* wrow = (nrow < PROJ) ? (W_k + (long)nrow * HIDDEN)
                                      : (W_v + (long)(nrow - PROJ) * HIDDEN);

    v8f acc[4] = {};
    #pragma unroll 2
    for (int kk = 0; kk < HIDDEN; kk += 4) {
        v2f b;
        b.x = wrow[kk + kb];
        b.y = wrow[kk + kb + 1];
        #pragma unroll
        for (int rt = 0; rt < 4; ++rt) {
            v2f a;
            const float* arow = xs + (rt * 16 + mn) * XSTRIDE;
            a.x = arow[kk + kb];
            a.y = arow[kk + kb + 1];
            acc[rt] = __builtin_amdgcn_wmma_f32_16x16x4_f32(
                          false, a, false, b, (short)0, acc[rt], false, false);
        }
    }

    // Epilogue: store proj; for key columns also atomically accumulate ksum.
    // D layout: VGPR r -> row (rt*16 + r + mh), col (n0 + mn).
    const bool isKeyCols = (n0 < PROJ);          // wave-uniform
    if (e0 + RPB <= nnz) {                       // full-tile fast path
        #pragma unroll
        for (int rt = 0; rt < 4; ++rt)
            #pragma unroll
            for (int r = 0; r < 8; ++r) {
                const int e = e0 + rt * 16 + r + mh;
                const float v = acc[rt][r];
                proj[(long)e * PROJ2 + n0 + mn] = v;
                if (isKeyCols)
                    atomicAdd(&ksum[(long)he_idx[e] * PROJ + n0 + mn], v);
            }
    } else {
        #pragma unroll
        for (int rt = 0; rt < 4; ++rt)
            #pragma unroll
            for (int r = 0; r < 8; ++r) {
                const int e = e0 + rt * 16 + r + mh;
                if (e < nnz) {
                    const float v = acc[rt][r];
                    proj[(long)e * PROJ2 + n0 + mn] = v;
                    if (isKeyCols)
                        atomicAdd(&ksum[(long)he_idx[e] * PROJ + n0 + mn], v);
                }
            }
    }
}

// ---------------------------------------------------------------------------
// K2: distance-to-centroid score + segment max (one wave32 per edge).
// ---------------------------------------------------------------------------
__global__ __launch_bounds__(256)
void k_scores(const float* __restrict__ proj,
              const int* __restrict__ he_idx,
              const float* __restrict__ ksum, const float* __restrict__ cnt,
              float* __restrict__ scores, float* __restrict__ smax, int nnz)
{
    const int e = blockIdx.x * 8 + (threadIdx.x >> 5);
    if (e >= nnz) return;
    const int lane = threadIdx.x & 31;
    const int he = he_idx[e];
    const float inv = 1.0f / fmaxf(cnt[he], 1.0f);

    float ss = 0.0f;
    #pragma unroll
    for (int j = 0; j < 2; ++j) {
        const int p = lane + j * 32;
        const float d = proj[(long)e * PROJ2 + p]
                      - ksum[(long)he * PROJ + p] * inv;
        ss += d * d;
    }
    #pragma unroll
    for (int off = 16; off > 0; off >>= 1)
        ss += __shfl_down(ss, off, 32);

    if (lane == 0) {
        const float score = sqrtf(ss) * 0.125f;  // scale = 1/sqrt(64)
        scores[e] = score;
        // score >= 0, so int-ordered atomicMax == float max; smax zero-init
        // also reproduces the reference's isfinite() guard for empty segments.
        atomicMax((int*)&smax[he], __float_as_int(score));
    }
}

// ---------------------------------------------------------------------------
// K3: ex = exp(score - smax[he]); denominator segment-sum. scores reused as ex.
// ---------------------------------------------------------------------------
__global__ __launch_bounds__(256)
void k_exp(float* __restrict__ scores, const int* __restrict__ he_idx,
           const float* __restrict__ smax, float* __restrict__ denom, int nnz)
{
    const int e = blockIdx.x * blockDim.x + threadIdx.x;
    if (e >= nnz) return;
    const int he = he_idx[e];
    const float ex = expf(scores[e] - smax[he]);
    scores[e] = ex;
    atomicAdd(&denom[he], ex);
}

// ---------------------------------------------------------------------------
// K4: weighted segment aggregation of vals.
// ---------------------------------------------------------------------------
__global__ __launch_bounds__(256)
void k_aggregate(const float* __restrict__ proj,
                 const float* __restrict__ scores,
                 const int* __restrict__ he_idx,
                 const float* __restrict__ denom,
                 float* __restrict__ hefeats, int nnz)
{
    const long idx = (long)blockIdx.x * blockDim.x + threadIdx.x;
    if (idx >= (long)nnz * PROJ) return;
    const int e  = (int)(idx >> 6);
    const int p  = (int)(idx & 63);
    const int he = he_idx[e];
    const float attn = scores[e] / fmaxf(denom[he], 1e-30f);
    atomicAdd(&hefeats[(long)he * PROJ + p],
              attn * proj[(long)e * PROJ2 + PROJ + p]);
}

// ---------------------------------------------------------------------------
// K5: out = he_feats @ W_v  ([HE,64] x [64,256]).  WMMA f32 16x16x4.
// One block = 64 rows; 8 waves x 2 column subtiles, 4 row-tile accumulators
// so each W_v fragment feeds 4 WMMAs.
// ---------------------------------------------------------------------------
__global__ __launch_bounds__(256)
void k_out_gemm(const float* __restrict__ hefeats,
                const float* __restrict__ W_v,
                float* __restrict__ out, int num_he)
{
    __shared__ float hs[RPB * HSTRIDE];   // ~17 KB
    const int e0 = blockIdx.x * RPB;
    const int t  = threadIdx.x;
    {
        const int r = t >> 2;
        const int c = (t & 3) * 16;
        const int e = e0 + r;
        #pragma unroll
        for (int i = 0; i < 16; i += 4) {
            float4 v = (e < num_he)
                     ? *(const float4*)(hefeats + (long)e * PROJ + c + i)
                     : make_float4(0.f, 0.f, 0.f, 0.f);
            *(float4*)(hs + r * HSTRIDE + c + i) = v;
        }
    }
    __syncthreads();

    const int wave = t >> 5;
    const int lane = t & 31;
    const int mn   = lane & 15;
    const int kb   = (lane >> 4) * 2;
    const int mh   = (lane >> 4) * 8;
    const bool full = (e0 + RPB) <= num_he;

    #pragma unroll
    for (int sub = 0; sub < 2; ++sub) {
        const int n0 = wave * 32 + sub * 16;
        v8f acc[4] = {};
        #pragma unroll 2
        for (int kk = 0; kk < PROJ; kk += 4) {
            v2f b;                                   // B[k][n] = W_v[k][n]
            b.x = W_v[(long)(kk + kb)     * HIDDEN + n0 + mn];
            b.y = W_v[(long)(kk + kb + 1) * HIDDEN + n0 + mn];
            #pragma unroll
            for (int rt = 0; rt < 4; ++rt) {
                v2f a;
                const float* arow = hs + (rt * 16 + mn) * HSTRIDE;
                a.x = arow[kk + kb];
                a.y = arow[kk + kb + 1];
                acc[rt] = __builtin_amdgcn_wmma_f32_16x16x4_f32(
                              false, a, false, b, (short)0, acc[rt], false, false);
            }
        }
        if (full) {
            #pragma unroll
            for (int rt = 0; rt < 4; ++rt)
                #pragma unroll
                for (int r = 0; r < 8; ++r) {
                    const int e = e0 + rt * 16 + r + mh;
                    out[(long)e * HIDDEN + n0 + mn] = acc[rt][r];
                }
        } else {
            #pragma unroll
            for (int rt = 0; rt < 4; ++rt)
                #pragma unroll
                for (int r = 0; r < 8; ++r) {
                    const int e = e0 + rt * 16 + r + mh;
                    if (e < num_he) out[(long)e * HIDDEN + n0 + mn] = acc[rt][r];
                }
        }
    }
}

// ---------------------------------------------------------------------------
__global__ __launch_bounds__(256)
void k_zero(float* __restrict__ p, long n)
{
    const long i = (long)blockIdx.x * blockDim.x + threadIdx.x;
    if (i < n) p[i] = 0.0f;
}

extern "C" void kernel_launch(void* const* d_in, const int* in_sizes, int n_in,
                              void* d_out, int out_size, void* d_ws, size_t ws_size,
                              hipStream_t stream)
{
    const float* node_feats = (const float*)d_in[0];
    const float* W_k        = (const float*)d_in[1];
    const float* W_v        = (const float*)d_in[2];
    const int*   node_idx   = (const int*)d_in[3];
    const int*   he_idx     = (const int*)d_in[4];
    (void)n_in; (void)ws_size;

    const int nnz = in_sizes[3];
    const int HE  = out_size / HIDDEN;      // num_hyperedges

    // Workspace layout (f32). Zeroed region is contiguous: ksum..denom.
    float* ws = (float*)d_ws;
    long o = 0;
    float* proj    = ws + o; o += (long)nnz * PROJ2;   // keys|vals
    float* ksum    = ws + o; o += (long)HE * PROJ;
    float* hefeats = ws + o; o += (long)HE * PROJ;
    float* cnt     = ws + o; o += HE;
    float* smax    = ws + o; o += HE;
    float* denom   = ws + o; o += HE;
    float* scores  = ws + o; o += nnz;

    const long zn = (long)HE * PROJ * 2 + 3L * HE;     // ksum,hefeats,cnt,smax,denom
    k_zero<<<(int)((zn + 255) / 256), 256, 0, stream>>>(ksum, zn);

    k_project<<<(nnz + RPB - 1) / RPB, 256, 0, stream>>>(
        node_feats, W_k, W_v, node_idx, he_idx, proj, ksum, cnt, nnz);

    k_scores<<<(nnz + 7) / 8, 256, 0, stream>>>(
        proj, he_idx, ksum, cnt, scores, smax, nnz);

    k_exp<<<(nnz + 255) / 256, 256, 0, stream>>>(
        scores, he_idx, smax, denom, nnz);

    const long t2 = (long)nnz * PROJ;
    k_aggregate<<<(int)((t2 + 255) / 256), 256, 0, stream>>>(
        proj, scores, he_idx, denom, hefeats, nnz);

    k_out_gemm<<<(HE + RPB - 1) / RPB, 256, 0, stream>>>(
        hefeats, W_v, (float*)d_out, HE);
}